// EllipsoidSampler_3375844295212
// MI455X (gfx1250) — compile-verified
//
#include <hip/hip_runtime.h>
#include <stdint.h>

// EllipsoidSampler for MI455X (gfx1250, wave32).
//
// Roofline: naive = 340 MB noise reads (~16 us @ 23.3 TB/s). Bounding-box
// guard (margin = 8 sigma of N(0,0.5) noise -> exact in practice) cuts noise
// traffic to ~5 MB; total ~34 MB => ~1.5 us memory floor.
// Secondary bottleneck is the per-thread unrolled miss-test (12 ellipsoids
// for every voxel). We keep it to ~6 VALU per ellipsoid by holding all
// centers/axes in SGPRs (uniform s_load) and testing with |abs|-modifier
// float compares, and we use vec-8 voxels/thread to halve the wave count.
// CDNA5-specific paths:
//   * global_load_async_to_lds_b32 + s_wait_asynccnt : DMA the 12 labels
//     into LDS; consumed via broadcast ds_load only inside hit blocks.
//   * global_prefetch_b8 : warm next ellipsoid's noise line on a hit.
// WMMA is inapplicable (no matmul structure).

#define DD 192
#define HH 192
#define WW 192
#define EE 12
#define HWC (HH * WW)
#define DHW (DD * HH * WW)
#define VEC 8                       // voxels per thread (2 x float4)
#define TPR (WW / VEC)              // 24 threads per x-row
#define MARGIN 4.0f                 // 8 sigma of N(0,0.5)

__global__ __launch_bounds__(256) void EllipsoidSampler_kernel(
    const int*   __restrict__ centers,  // [E,3]
    const int*   __restrict__ axes,     // [E,3]
    const float* __restrict__ noise,    // [E,D,H,W]
    const float* __restrict__ labels,   // [E]
    float*       __restrict__ out)      // [D,H,W]
{
    // ---- Stage the 12 labels into LDS with the gfx1250 async global->LDS DMA.
    __shared__ float s_lab[16];
    const int tid = threadIdx.x;
    if (tid < EE) {
        unsigned lds_off = (unsigned)(size_t)(&s_lab[tid]);   // LDS byte offset
        uint64_t gaddr   = (uint64_t)(size_t)(labels + tid);
        asm volatile("global_load_async_to_lds_b32 %0, %1, off"
                     :: "v"(lds_off), "v"(gaddr) : "memory");
    }
    asm volatile("s_wait_asynccnt 0" ::: "memory");
    __syncthreads();

    // ---- Each thread owns 8 contiguous voxels along W.
    const int i  = blockIdx.x * 256 + tid;        // 0 .. 192*192*24-1
    const int xt = (i % TPR) * VEC;
    const int y  = (i / TPR) % HH;
    const int z  = i / (TPR * HH);
    const int base = z * HWC + y * WW + xt;

    const float zf  = (float)z;
    const float yf  = (float)y;
    const float xlo = (float)xt;
    const float xhi = (float)(xt + VEC - 1);

    float v[VEC];
#pragma unroll
    for (int j = 0; j < VEC; ++j) v[j] = 0.0f;

#pragma unroll
    for (int e = 0; e < EE; ++e) {
        // Uniform-index loads -> SGPR-resident constants (s_load + one-time cvt).
        const float cz = (float)centers[e * 3 + 0];
        const float cy = (float)centers[e * 3 + 1];
        const float cx = (float)centers[e * 3 + 2];
        const float az = (float)axes[e * 3 + 0];
        const float ay = (float)axes[e * 3 + 1];
        const float ax = (float)axes[e * 3 + 2];

        const float dz = zf - cz;                 // reused in hit block
        const float dy = yf - cy;

        // q<=1 requires |d| <= a+n per axis; |n|<=4 w.p. 1-1e-15 per sample.
        // Single fused condition: ~6 v_cmp/v_sub (abs is a free src modifier).
        const bool hit =
            (fabsf(dz) <= az + MARGIN) &
            (fabsf(dy) <= ay + MARGIN) &
            (xhi >= cx - (ax + MARGIN)) &
            (xlo <= cx + (ax + MARGIN));

        if (hit) {  // wave-coherent -> s_cbranch_execz skips all VMEM outside bbox
            const float* np = noise + (size_t)e * DHW + base;
            const float4 n0 = *(const float4*)(np);
            const float4 n1 = *(const float4*)(np + 4);
            if (e + 1 < EE)  // overlapping bboxes: warm cache for next ellipsoid
                __builtin_prefetch(noise + (size_t)(e + 1) * DHW + base, 0, 1);

            const float lab = s_lab[e];           // ds_load_b32 broadcast (async-staged)
            const float nn[VEC] = {n0.x, n0.y, n0.z, n0.w, n1.x, n1.y, n1.z, n1.w};
#pragma unroll
            for (int j = 0; j < VEC; ++j) {
                const float dx = (float)(xt + j) - cx;
                const float t0 = dz / (az + nn[j]);   // match reference: div, then square
                const float t1 = dy / (ay + nn[j]);
                const float t2 = dx / (ax + nn[j]);
                const float q  = t0 * t0 + t1 * t1 + t2 * t2;
                if (q <= 1.0f) v[j] = lab;            // later e overwrites earlier
            }
        }
    }

    // Every voxel written exactly once (two b128 stores).
    float4* o = (float4*)(out + base);
    o[0] = make_float4(v[0], v[1], v[2], v[3]);
    o[1] = make_float4(v[4], v[5], v[6], v[7]);
}

extern "C" void kernel_launch(void* const* d_in, const int* in_sizes, int n_in,
                              void* d_out, int out_size, void* d_ws, size_t ws_size,
                              hipStream_t stream) {
    (void)in_sizes; (void)n_in; (void)d_ws; (void)ws_size; (void)out_size;
    const int*   centers = (const int*)d_in[0];    // [12,3] int32
    const int*   axes    = (const int*)d_in[1];    // [12,3] int32
    const float* noise   = (const float*)d_in[2];  // [12,192,192,192] f32
    const float* labels  = (const float*)d_in[3];  // [12] f32
    float*       out     = (float*)d_out;          // [192,192,192] f32

    const int total_threads = DD * HH * TPR;       // 884,736
    const int blocks = total_threads / 256;        // 3456
    EllipsoidSampler_kernel<<<blocks, 256, 0, stream>>>(centers, axes, noise, labels, out);
}